// MyModel_61933428413835
// MI455X (gfx1250) — compile-verified
//
#include <hip/hip_runtime.h>

// COO scatter-add: out[i0,i1] += v, out is 2048x2048 f32.
// Bandwidth-bound (~192 MB streamed @ 23.3 TB/s -> ~8us floor). gfx1250 choices:
//  - b128 non-temporal loads for the read-once streams (protect L2-resident 16MB output)
//  - native no-return global_atomic_add_f32 (unsafeAtomicAdd), STOREcnt-tracked
//  - 2x unrolled grid-stride loop: 6 independent b128 loads in flight per iter

typedef float f32x4 __attribute__((ext_vector_type(4)));
typedef int   i32x4 __attribute__((ext_vector_type(4)));

__global__ void __launch_bounds__(256)
scatter_add_kernel(const float* __restrict__ vals,
                   const int* __restrict__ row,   // indices[0]
                   const int* __restrict__ col,   // indices[1]
                   float* __restrict__ out,
                   int nnz, int n8) {
    int t = blockIdx.x * blockDim.x + threadIdx.x;
    int stride = gridDim.x * blockDim.x;

    const f32x4* __restrict__ v4 = (const f32x4*)vals;
    const i32x4* __restrict__ r4 = (const i32x4*)row;
    const i32x4* __restrict__ c4 = (const i32x4*)col;

    // Main loop: 8 nonzeros per thread per iteration.
    // 6x b128 NT loads issued back-to-back, then 8 no-return f32 atomics.
    for (int i = t; i < n8; i += stride) {
        int j = i * 2;
        f32x4 v0 = __builtin_nontemporal_load(v4 + j);
        f32x4 v1 = __builtin_nontemporal_load(v4 + j + 1);
        i32x4 r0 = __builtin_nontemporal_load(r4 + j);
        i32x4 r1 = __builtin_nontemporal_load(r4 + j + 1);
        i32x4 c0 = __builtin_nontemporal_load(c4 + j);
        i32x4 c1 = __builtin_nontemporal_load(c4 + j + 1);
#pragma unroll
        for (int k = 0; k < 4; ++k) {
            // D == 2048, col in [0,2048): r*2048 + c == (r<<11)|c
            unsigned off = ((unsigned)r0[k] << 11) | (unsigned)c0[k];
            unsafeAtomicAdd(out + off, v0[k]);
        }
#pragma unroll
        for (int k = 0; k < 4; ++k) {
            unsigned off = ((unsigned)r1[k] << 11) | (unsigned)c1[k];
            unsafeAtomicAdd(out + off, v1[k]);
        }
    }

    // Scalar tail (nnz not divisible by 8; no-op for NNZ = 2^24).
    int base = n8 * 8;
    for (int i = base + t; i < nnz; i += stride) {
        unsigned off = ((unsigned)row[i] << 11) | (unsigned)col[i];
        unsafeAtomicAdd(out + off, vals[i]);
    }
}

__global__ void __launch_bounds__(256)
zero_out_kernel(float* __restrict__ out, int n4) {
    int t = blockIdx.x * blockDim.x + threadIdx.x;
    int stride = gridDim.x * blockDim.x;
    f32x4 z = {0.0f, 0.0f, 0.0f, 0.0f};
    for (int i = t; i < n4; i += stride) {
        ((f32x4* __restrict__)out)[i] = z;
    }
}

extern "C" void kernel_launch(void* const* d_in, const int* in_sizes, int n_in,
                              void* d_out, int out_size, void* d_ws, size_t ws_size,
                              hipStream_t stream) {
    const float* vals = (const float*)d_in[0];
    const int*   idx  = (const int*)d_in[1];   // shape (3, NNZ) row-major
    int nnz = in_sizes[0];

    const int* row = idx;           // indices[0]
    const int* col = idx + nnz;     // indices[1]  (indices[2] unused by the op)

    float* out = (float*)d_out;     // 2048*2048 f32

    // Zero the accumulator first (stream-ordered before the scatter).
    int n4out = out_size / 4;
    zero_out_kernel<<<2048, 256, 0, stream>>>(out, n4out);

    // Scatter-add: 4096 blocks x 256 threads (8 wave32 per block), grid-stride,
    // 8 nonzeros per thread per iteration -> exactly 2 iterations per thread.
    int n8 = nnz / 8;
    scatter_add_kernel<<<4096, 256, 0, stream>>>(vals, row, col, out, nnz, n8);
}